// GraphEncoder_68195490725960
// MI455X (gfx1250) — compile-verified
//
#include <hip/hip_runtime.h>
#include <hip/hip_bf16.h>

// ---------------------------------------------------------------------------
// GAT (2-layer, H=4 heads, C=32) for gfx1250.
// GEMMs use V_WMMA_F32_16X16X4_F32 (full fp32 matrix pipe).
// Edge phase uses native f32 global atomics (L2-resident on 192MB L2).
// ---------------------------------------------------------------------------

typedef __attribute__((ext_vector_type(2))) float v2f;
typedef __attribute__((ext_vector_type(8))) float v8f;

#define HEADS 4
#define CPH   32
#define F     128   // = HEADS*CPH, also in_channels of both layers
#define SLOPE 0.2f

// ---------------- WMMA fp32 GEMM: out[n x 128] = A[n x 128] @ B[128 x 128] (+bias) ----
// One wave computes a 16-row x 128-col strip (8 accumulators of 16x16).
// A-fragment (16x4 f32, ISA 7.12.2): lane m = lane&15; VGPR0 holds K+0 (lanes 0-15)
// and K+2 (lanes 16-31); VGPR1 holds K+1 / K+3.
// B-fragment (4x16): row K striped across lanes, same half-wave K split.
// C/D (16x16 f32): acc[i] -> row = r0 + i + (lane>=16 ? 8 : 0), col = c0 + (lane&15).
__global__ void gat_gemm_wmma_f32(const float* __restrict__ A,
                                  const float* __restrict__ B,
                                  const float* __restrict__ bias, // nullable
                                  float* __restrict__ out,
                                  int n_rows) {
  const int lane  = threadIdx.x & 31;
  const int wave  = threadIdx.x >> 5;
  const int strip = blockIdx.x * 8 + wave;      // 16-row strip index
  if (strip * 16 >= n_rows) return;             // wave-uniform exit

  const int r0   = strip * 16;
  const int m    = r0 + (lane & 15);
  const int koff = (lane >> 4) ? 2 : 0;
  const int nb   = lane & 15;

  v8f acc[8];
  const v8f vzero = {0.f, 0.f, 0.f, 0.f, 0.f, 0.f, 0.f, 0.f};
#pragma unroll
  for (int t = 0; t < 8; ++t) acc[t] = vzero;

  for (int k = 0; k < F; k += 4) {
    v2f a;
    a.x = A[(size_t)m * F + k + koff];
    a.y = A[(size_t)m * F + k + koff + 1];
#pragma unroll
    for (int t = 0; t < 8; ++t) {
      const int n = t * 16 + nb;
      v2f b;
      b.x = B[(size_t)(k + koff) * F + n];
      b.y = B[(size_t)(k + koff + 1) * F + n];
      acc[t] = __builtin_amdgcn_wmma_f32_16x16x4_f32(
          /*neg_a=*/false, a, /*neg_b=*/false, b,
          /*c_mod=*/(short)0, acc[t], /*reuse_a=*/false, /*reuse_b=*/false);
    }
  }

  const int rbase = r0 + ((lane >> 4) ? 8 : 0);
#pragma unroll
  for (int t = 0; t < 8; ++t) {
    const int c = t * 16 + nb;
    const float bv = bias ? bias[c] : 0.f;
#pragma unroll
    for (int i = 0; i < 8; ++i) {
      out[(size_t)(rbase + i) * F + c] = acc[t][i] + bv;
    }
  }
}

// ---------------- attention coefficients: a_s[n,h]=sum_c xl[n,h,c]*att_src[h,c] -------
__global__ void gat_attn_coef(const float* __restrict__ xl,
                              const float* __restrict__ att_src,
                              const float* __restrict__ att_dst,
                              float* __restrict__ a_s,
                              float* __restrict__ a_d, int n) {
  const int idx = blockIdx.x * blockDim.x + threadIdx.x;
  if (idx >= n * HEADS) return;
  const int h    = idx & (HEADS - 1);
  const int node = idx >> 2;
  const float* v = xl + (size_t)node * F + h * CPH;
  float ss = 0.f, sd = 0.f;
#pragma unroll
  for (int c = 0; c < CPH; ++c) {
    const float x = v[c];
    ss += x * att_src[h * CPH + c];
    sd += x * att_dst[h * CPH + c];
  }
  a_s[idx] = ss;
  a_d[idx] = sd;
}

// ---------------- buffer init -------------------------------------------------------
__global__ void gat_fill_bias(float* __restrict__ out, const float* __restrict__ bias,
                              int total) {  // total = n*F
  const int idx = blockIdx.x * blockDim.x + threadIdx.x;
  if (idx < total) out[idx] = bias[idx & (F - 1)];
}

__global__ void gat_init_mz(unsigned* __restrict__ m_u, float* __restrict__ z, int total) {
  const int idx = blockIdx.x * blockDim.x + threadIdx.x;
  if (idx < total) { m_u[idx] = 0u; z[idx] = 0.f; }
}

// ordered-uint encoding of float for atomicMax (monotone in f)
__device__ __forceinline__ unsigned f2ord(float f) {
  unsigned u = __float_as_uint(f);
  return u ^ ((u & 0x80000000u) ? 0xFFFFFFFFu : 0x80000000u);
}
__device__ __forceinline__ float ord2f(unsigned u) {
  unsigned bits = (u & 0x80000000u) ? (u ^ 0x80000000u) : ~u;
  return __uint_as_float(bits);
}

__device__ __forceinline__ void edge_endpoints(int e, int E,
                                               const int* __restrict__ src,
                                               const int* __restrict__ dst,
                                               int& s, int& d) {
  if (e < E) { s = src[e]; d = dst[e]; }
  else       { s = d = e - E; }        // self loop
}

// ---------------- pass 1: leaky-relu logits + segment max (atomicMax on ordered uint) -
__global__ void gat_edge_logits_max(const int* __restrict__ src,
                                    const int* __restrict__ dst,
                                    const float* __restrict__ a_s,
                                    const float* __restrict__ a_d,
                                    float* __restrict__ ev,
                                    unsigned* __restrict__ m_u,
                                    int E, int n) {
  const int e = blockIdx.x * blockDim.x + threadIdx.x;
  if (e >= E + n) return;
  int s, d;
  edge_endpoints(e, E, src, dst, s, d);
#pragma unroll
  for (int h = 0; h < HEADS; ++h) {
    float v = a_s[s * HEADS + h] + a_d[d * HEADS + h];
    v = (v > 0.f) ? v : SLOPE * v;
    ev[(size_t)e * HEADS + h] = v;
    atomicMax(&m_u[d * HEADS + h], f2ord(v));
  }
}

// ---------------- pass 2: p = exp(e - m[dst]); segment sum --------------------------
__global__ void gat_edge_expsum(const int* __restrict__ src,
                                const int* __restrict__ dst,
                                float* __restrict__ ev,       // in: logits, out: p
                                const unsigned* __restrict__ m_u,
                                float* __restrict__ z,
                                int E, int n) {
  const int idx = blockIdx.x * blockDim.x + threadIdx.x;
  if (idx >= (E + n) * HEADS) return;
  const int e = idx >> 2;
  const int h = idx & (HEADS - 1);
  int s, d;
  edge_endpoints(e, E, src, dst, s, d);
  (void)s;
  const float m = ord2f(m_u[d * HEADS + h]);
  const float p = expf(ev[idx] - m);
  ev[idx] = p;
  unsafeAtomicAdd(&z[d * HEADS + h], p);
}

// ---------------- pass 3: out[dst] += alpha * xl[src] -------------------------------
// One thread per (edge, feature); j fast so xl row reads are coalesced and the
// atomic adds per edge hit 2 consecutive cachelines.
__global__ void gat_edge_aggregate(const int* __restrict__ src,
                                   const int* __restrict__ dst,
                                   const float* __restrict__ xl,
                                   const float* __restrict__ p,
                                   const float* __restrict__ z,
                                   float* __restrict__ out,
                                   int E, int n) {
  const long long idx = (long long)blockIdx.x * blockDim.x + threadIdx.x;
  const long long total = (long long)(E + n) * F;
  if (idx >= total) return;
  const int e = (int)(idx >> 7);
  const int j = (int)(idx & (F - 1));
  const int h = j >> 5;
  int s, d;
  edge_endpoints(e, E, src, dst, s, d);
  const float alpha = p[(size_t)e * HEADS + h] / z[d * HEADS + h];
  unsafeAtomicAdd(&out[(size_t)d * F + j], xl[(size_t)s * F + j] * alpha);
}

// ---------------------------------------------------------------------------
extern "C" void kernel_launch(void* const* d_in, const int* in_sizes, int n_in,
                              void* d_out, int out_size, void* d_ws, size_t ws_size,
                              hipStream_t stream) {
  const float* x        = (const float*)d_in[0];
  const int*   eidx     = (const int*)d_in[1];
  const float* W1       = (const float*)d_in[2];
  const float* att_s1   = (const float*)d_in[3];
  const float* att_d1   = (const float*)d_in[4];
  const float* b1       = (const float*)d_in[5];
  const float* W2       = (const float*)d_in[6];
  const float* att_s2   = (const float*)d_in[7];
  const float* att_d2   = (const float*)d_in[8];
  const float* b2       = (const float*)d_in[9];
  const float* res_W2   = (const float*)d_in[10];

  const int N    = in_sizes[0] / F;       // 50000
  const int E    = in_sizes[1] / 2;       // 800000
  const int Etot = E + N;
  const int* src = eidx;
  const int* dst = eidx + E;

  // workspace carve-out
  float*    xl  = (float*)d_ws;                 // N*F   (reused for both layers)
  float*    h1  = xl  + (size_t)N * F;          // N*F   (layer-1 output)
  float*    a_s = h1  + (size_t)N * F;          // N*H
  float*    a_d = a_s + (size_t)N * HEADS;      // N*H
  unsigned* m_u = (unsigned*)(a_d + (size_t)N * HEADS); // N*H
  float*    z   = (float*)(m_u + (size_t)N * HEADS);    // N*H
  float*    ev  = z + (size_t)N * HEADS;        // Etot*H (logits, then p)
  float*    out = (float*)d_out;                // N*F

  const int TB = 256;
  const int gemm_blocks = ((N + 15) / 16 + 7) / 8;
  const int nh_blocks   = (N * HEADS + TB - 1) / TB;
  const int nf_blocks   = (N * F + TB - 1) / TB;
  const int e_blocks    = (Etot + TB - 1) / TB;
  const int eh_blocks   = (Etot * HEADS + TB - 1) / TB;
  const long long agg_total = (long long)Etot * F;
  const int agg_blocks  = (int)((agg_total + TB - 1) / TB);

  // ---------------- layer 1 ----------------
  gat_gemm_wmma_f32<<<gemm_blocks, TB, 0, stream>>>(x, W1, nullptr, xl, N);
  gat_attn_coef<<<nh_blocks, TB, 0, stream>>>(xl, att_s1, att_d1, a_s, a_d, N);
  gat_init_mz<<<nh_blocks, TB, 0, stream>>>(m_u, z, N * HEADS);
  gat_fill_bias<<<nf_blocks, TB, 0, stream>>>(h1, b1, N * F);
  gat_edge_logits_max<<<e_blocks, TB, 0, stream>>>(src, dst, a_s, a_d, ev, m_u, E, N);
  gat_edge_expsum<<<eh_blocks, TB, 0, stream>>>(src, dst, ev, m_u, z, E, N);
  gat_edge_aggregate<<<agg_blocks, TB, 0, stream>>>(src, dst, xl, ev, z, h1, E, N);

  // ---------------- layer 2 ----------------
  gat_gemm_wmma_f32<<<gemm_blocks, TB, 0, stream>>>(h1, W2, nullptr, xl, N);
  gat_attn_coef<<<nh_blocks, TB, 0, stream>>>(xl, att_s2, att_d2, a_s, a_d, N);
  gat_init_mz<<<nh_blocks, TB, 0, stream>>>(m_u, z, N * HEADS);
  // d_out = h1 @ res_W2 + b2  (residual + bias), aggregation accumulates on top
  gat_gemm_wmma_f32<<<gemm_blocks, TB, 0, stream>>>(h1, res_W2, b2, out, N);
  gat_edge_logits_max<<<e_blocks, TB, 0, stream>>>(src, dst, a_s, a_d, ev, m_u, E, N);
  gat_edge_expsum<<<eh_blocks, TB, 0, stream>>>(src, dst, ev, m_u, z, E, N);
  gat_edge_aggregate<<<agg_blocks, TB, 0, stream>>>(src, dst, xl, ev, z, out, E, N);

  (void)n_in; (void)out_size; (void)ws_size;
}